// GDE_4939212390990
// MI455X (gfx1250) — compile-verified
//
#include <hip/hip_runtime.h>

#define NC      6
#define NF      2096
#define NB      1024
#define TILE_M  32                 // rows per block = 2 WMMA M-tiles
#define JSPLIT  4                  // split of output columns across blocks
#define JCHUNKS (NF / 16)          // 131 chunks of 16 columns
#define NWAVES  8                  // 256 threads / wave32

typedef __attribute__((ext_vector_type(2))) float v2f;
typedef __attribute__((ext_vector_type(8))) float v8f;

__global__ void GDE_zero_out(float* out) {
    int i = blockIdx.x * blockDim.x + threadIdx.x;
    if (i < NB) out[i] = 0.0f;
}

__global__ __launch_bounds__(256)
void GDE_mahal_wmma(const float* __restrict__ X,
                    const float* __restrict__ mean,
                    const float* __restrict__ inv_cov,
                    const int*   __restrict__ labels,
                    float*       __restrict__ out) {
    extern __shared__ float smem[];
    float* sdelta = smem;                       // TILE_M * NF floats (delta tile)
    float* qsum   = smem + TILE_M * NF;         // TILE_M floats (per-row reduction)
    int*   slab   = (int*)(qsum + TILE_M);      // TILE_M labels

    const int tid  = threadIdx.x;
    const int lane = tid & 31;
    const int wave = tid >> 5;
    const int ln15 = lane & 15;                 // A: row M; B/C: column N
    const int half = lane >> 4;                 // K-pair / M-half selector
    const int row0 = blockIdx.x * TILE_M;

    // j-chunk slice for this block
    const int chunksPer = (JCHUNKS + JSPLIT - 1) / JSPLIT;   // 33
    const int cb = blockIdx.y * chunksPer;
    const int ce = (cb + chunksPer < JCHUNKS) ? (cb + chunksPer) : JCHUNKS;

    if (tid < TILE_M) slab[tid] = labels[row0 + tid];
    __syncthreads();

    // Labels are sorted -> distinct classes in this tile are adjacent runs.
    for (int t = 0; t < TILE_M; ++t) {
        const int c = slab[t];
        if (t > 0 && c == slab[t - 1]) continue;   // uniform: slab is in LDS

        // ---- Stage delta tile (X - mean[c]) into LDS ----
        for (int idx = tid; idx < TILE_M * NF; idx += 256) {
            int r   = idx / NF;
            int col = idx - r * NF;
            sdelta[idx] = X[(size_t)(row0 + r) * NF + col]
                        - mean[(size_t)c * NF + col];
        }
        if (tid < TILE_M) qsum[tid] = 0.0f;
        __syncthreads();

        const float* ic = inv_cov + (size_t)c * NF * NF;

        float qp0[8], qp1[8];
        #pragma unroll
        for (int r = 0; r < 8; ++r) { qp0[r] = 0.0f; qp1[r] = 0.0f; }

        // ---- GEMM-like loop: t = delta @ inv_cov[c], fused with q += t*delta ----
        for (int jc = cb + wave; jc < ce; jc += NWAVES) {
            const int j0 = jc * 16;
            v8f acc0 = {};   // rows 0..15 of tile
            v8f acc1 = {};   // rows 16..31 of tile

            // A layout (16x4 f32): lanes 0-15 hold M=ln15, K={0,1}; lanes 16-31 K={2,3}
            const float* sa0 = sdelta + (size_t)ln15 * NF + 2 * half;
            const float* sa1 = sa0 + (size_t)16 * NF;
            // B layout (4x16 f32): v0 = rows K={0,2}, v1 = rows K={1,3}; N = ln15
            const float* bp  = ic + (size_t)(2 * half) * NF + j0 + ln15;

            for (int i0 = 0; i0 < NF; i0 += 4) {
                v2f a0 = *(const v2f*)(sa0 + i0);
                v2f a1 = *(const v2f*)(sa1 + i0);
                v2f b;
                b.x = bp[(size_t)i0 * NF];
                b.y = bp[(size_t)i0 * NF + NF];
                acc0 = __builtin_amdgcn_wmma_f32_16x16x4_f32(
                           false, a0, false, b, (short)0, acc0, false, false);
                acc1 = __builtin_amdgcn_wmma_f32_16x16x4_f32(
                           false, a1, false, b, (short)0, acc1, false, false);
            }

            // C/D layout: VGPR r holds row M = 8*half + r, column N = j0 + ln15
            #pragma unroll
            for (int r = 0; r < 8; ++r) {
                qp0[r] += acc0[r] * sdelta[(size_t)(8 * half + r) * NF + j0 + ln15];
                qp1[r] += acc1[r] * sdelta[(size_t)(16 + 8 * half + r) * NF + j0 + ln15];
            }
        }

        // ---- Reduce across lanes/waves via LDS float atomics (ds_add_f32) ----
        #pragma unroll
        for (int r = 0; r < 8; ++r) {
            atomicAdd(&qsum[8 * half + r],      qp0[r]);
            atomicAdd(&qsum[16 + 8 * half + r], qp1[r]);
        }
        __syncthreads();

        // ---- Accumulate this block's partial q into global output ----
        if (tid < TILE_M && slab[tid] == c) {
            atomicAdd(&out[row0 + tid], qsum[tid]);   // global_atomic_add_f32
        }
        __syncthreads();
    }
}

extern "C" void kernel_launch(void* const* d_in, const int* in_sizes, int n_in,
                              void* d_out, int out_size, void* d_ws, size_t ws_size,
                              hipStream_t stream) {
    const float* X       = (const float*)d_in[0];
    const float* mean    = (const float*)d_in[1];
    const float* inv_cov = (const float*)d_in[2];
    const int*   labels  = (const int*)d_in[3];
    float*       out     = (float*)d_out;

    GDE_zero_out<<<dim3((NB + 255) / 256), dim3(256), 0, stream>>>(out);

    size_t smem_bytes = (size_t)TILE_M * NF * sizeof(float)   // delta tile (~262 KB, fits 320 KB WGP LDS)
                      + TILE_M * sizeof(float)                // qsum
                      + TILE_M * sizeof(int);                 // labels
    dim3 grid(NB / TILE_M, JSPLIT);                           // 32 x 4 = 128 blocks
    GDE_mahal_wmma<<<grid, dim3(256), smem_bytes, stream>>>(X, mean, inv_cov, labels, out);
}